// CliffordMeanField3DLayerExact_13950053777568
// MI455X (gfx1250) — compile-verified
//
#include <hip/hip_runtime.h>

// MI455X / gfx1250: wave32, WMMA (V_WMMA_F32_16X16X4_F32) for the fp32 GEMMs.
// Bandwidth-bound workload (~768MB min HBM traffic @23.3TB/s): fuse everything,
// read x once into LDS per tile, write out once.

#define B_    8
#define N_    16384
#define DIM_  512
#define MV_   8
#define SCALE_ 0.1f
#define EPS_   1e-5f

#define ROWS_ 128          // rows per block (8 waves x 16)
#define XS_STRIDE 516      // 512 + 4 pad -> conflict-free 16-row column access

typedef __attribute__((ext_vector_type(2))) float v2f;
typedef __attribute__((ext_vector_type(8))) float v8f;

// ---------- K0: partial sums of x over N (split into 32 chunks, deterministic) ----------
__global__ void k_partial(const float* __restrict__ x, float* __restrict__ partial) {
    int s = blockIdx.x;          // 0..31
    int b = blockIdx.y;          // 0..7
    int tid = threadIdx.x;       // 256
    for (int d = tid; d < DIM_; d += 256) {
        float acc = 0.f;
        const float* p = x + ((size_t)b * N_ + (size_t)s * 512) * DIM_ + d;
        for (int n = 0; n < 512; ++n) acc += p[(size_t)n * DIM_];
        partial[((size_t)b * 32 + s) * DIM_ + d] = acc;
    }
}

// ---------- K1: finalize mean over N ----------
__global__ void k_mean(const float* __restrict__ partial, float* __restrict__ mean) {
    int idx = blockIdx.x * 256 + threadIdx.x;  // 0..4095
    int b = idx >> 9, d = idx & 511;
    float acc = 0.f;
    for (int s = 0; s < 32; ++s) acc += partial[((size_t)b * 32 + s) * DIM_ + d];
    mean[idx] = acc * (1.f / (float)N_);
}

// ---------- K2: t[b][m] = mean[b,:] . Wm[m,:] + bm[m] ----------
__global__ void k_t(const float* __restrict__ mean, const float* __restrict__ Wm,
                    const float* __restrict__ bm, float* __restrict__ t) {
    int tid = threadIdx.x;       // 64 threads, 1 block
    int b = tid >> 3, m = tid & 7;
    float acc = bm[m];
    for (int d = 0; d < DIM_; ++d) acc += mean[b * DIM_ + d] * Wm[m * DIM_ + d];
    t[b * 8 + m] = acc;
}

// Cl(3,0) Cayley: basis order 1,e1,e2,e3,e12,e13,e23,e123 (masks below).
// The idx<->mask map {0,1,2,4,3,5,6,7} is its own inverse (swaps 3<->4).
__device__ __forceinline__ void cayley_jk(int i, int j, int& k, float& sign) {
    const int masks[8] = {0, 1, 2, 4, 3, 5, 6, 7};
    int a = masks[i], bb = masks[j];
    int s = 0, aa = a >> 1;
    while (aa) { s ^= (__popc(aa & bb) & 1); aa >>= 1; }
    k = masks[a ^ bb];
    sign = s ? -1.f : 1.f;
}

// ---------- K3: W2[b][i][d] = sum_k (sum_j C[i][j][k]*t[b][j]) * Wo[d][k] ----------
__global__ void k_w2(const float* __restrict__ t, const float* __restrict__ Wo,
                     float* __restrict__ W2) {
    int idx = blockIdx.x * 256 + threadIdx.x;  // 0..32767
    int b = idx >> 12;
    int i = (idx >> 9) & 7;
    int d = idx & 511;
    float Mk[8] = {0, 0, 0, 0, 0, 0, 0, 0};
    float tb[8];
#pragma unroll
    for (int j = 0; j < 8; ++j) tb[j] = t[b * 8 + j];
#pragma unroll
    for (int j = 0; j < 8; ++j) {
        int k; float sg;
        cayley_jk(i, j, k, sg);
        Mk[k] += sg * tb[j];
    }
    float acc = 0.f;
#pragma unroll
    for (int k = 0; k < 8; ++k) acc += Mk[k] * Wo[d * 8 + k];
    W2[((size_t)b * 8 + i) * DIM_ + d] = acc;
}

// ---------- Main fused kernel ----------
// P = x_tile @ Wp^T + bp  (WMMA f32 16x16x4, K=512, MV padded 8->16 w/ zeros)
// out = P @ W2_b + bo ; y = x + 0.1*out ; LayerNorm over DIM
__global__ void k_main(const float* __restrict__ x, const float* __restrict__ Wp,
                       const float* __restrict__ bp, const float* __restrict__ W2,
                       const float* __restrict__ bo, const float* __restrict__ gamma,
                       const float* __restrict__ beta, float* __restrict__ out) {
    extern __shared__ float sm[];
    float* xs   = sm;            // 128*516 = 66048  (x tile, later y tile)
    float* wsh  = sm + 66048;    // 8192: Wp B-fragments (phase1), W2 B-fragments (phase2)
    float* pscr = sm + 74240;    // 8 waves * 16*18 = 2304: P relayout scratch
    float* bos  = sm + 76544;    // 512
    float* gs   = sm + 77056;    // 512
    float* bs   = sm + 77568;    // 512
    float* mus  = sm + 78080;    // 128
    float* rsds = sm + 78208;    // 128  -> total 78336 floats = 313344 B (< 320KB)

    const int tid = threadIdx.x;
    const int b = blockIdx.x >> 7;            // 128 blocks per batch
    const int n0 = (blockIdx.x & 127) * ROWS_;
    const float* xg = x + ((size_t)b * N_ + n0) * DIM_;

    // ---- stage x tile (float4, coalesced; LDS offsets 16B aligned since 516%4==0)
    for (int i = tid; i < ROWS_ * (DIM_ / 4); i += 256) {
        int r = i >> 7;
        int c4 = (i & 127) << 2;
        float4 v = *reinterpret_cast<const float4*>(xg + (size_t)r * DIM_ + c4);
        *reinterpret_cast<float4*>(&xs[r * XS_STRIDE + c4]) = v;
    }
    // ---- stage Wp^T in B-fragment order: [kk][half][n][pair], zero-padded n>=8
    for (int i = tid; i < 8192; i += 256) {
        int kk = i >> 6, rem = i & 63;
        int h = rem >> 5, n = (rem >> 1) & 15, p = rem & 1;
        int k = 4 * kk + 2 * h + p;
        wsh[i] = (n < 8) ? Wp[n * DIM_ + k] : 0.f;
    }
    for (int i = tid; i < DIM_; i += 256) { bos[i] = bo[i]; gs[i] = gamma[i]; bs[i] = beta[i]; }
    __syncthreads();

    const int wv = tid >> 5;
    const int lane = tid & 31;
    const int ncol = lane & 15;
    const int hh = lane >> 4;
    const int r0 = wv * 16;

    // ---- phase 1: P = x @ Wp^T + bp, 128 K-steps of v_wmma_f32_16x16x4_f32
    float bpn = (ncol < 8) ? bp[ncol] : 0.f;
    v8f acc = {bpn, bpn, bpn, bpn, bpn, bpn, bpn, bpn};
    const float* arow = &xs[(r0 + ncol) * XS_STRIDE + 2 * hh];
    for (int kk = 0; kk < 128; ++kk) {
        float2 av = *reinterpret_cast<const float2*>(arow + 4 * kk);
        float2 bv = *reinterpret_cast<const float2*>(&wsh[kk * 64 + hh * 32 + ncol * 2]);
        v2f a  = {av.x, av.y};
        v2f bf = {bv.x, bv.y};
        acc = __builtin_amdgcn_wmma_f32_16x16x4_f32(false, a, false, bf, (short)0, acc,
                                                    false, false);
    }
    // store P (C/D frag: VGPR i -> row i+8*hh, col ncol) to scratch for A-relayout
#pragma unroll
    for (int i = 0; i < 8; ++i)
        pscr[wv * 288 + (i + 8 * hh) * 18 + ncol] = acc[i];
    __syncthreads();

    // ---- stage W2_b B-fragments into wsh (overwrites Wp staging, now dead)
    for (int i = tid; i < 4096; i += 256) {
        int cc = i >> 7, rem = i & 127;
        int s = rem >> 6, h = (rem >> 5) & 1, n = (rem >> 1) & 15, p = rem & 1;
        int k = 4 * s + 2 * h + p;
        wsh[i] = W2[((size_t)b * 8 + k) * DIM_ + cc * 16 + n];
    }
    __syncthreads();

    // ---- phase 2: out = P @ W2 (K=8 -> 2 WMMA steps) fused with residual + stats
    float2 a0 = *reinterpret_cast<const float2*>(&pscr[wv * 288 + ncol * 18 + 2 * hh]);
    float2 a1 = *reinterpret_cast<const float2*>(&pscr[wv * 288 + ncol * 18 + 4 + 2 * hh]);
    v2f va0 = {a0.x, a0.y}, va1 = {a1.x, a1.y};
    float rsum[8] = {0, 0, 0, 0, 0, 0, 0, 0};
    float rsq[8]  = {0, 0, 0, 0, 0, 0, 0, 0};
    for (int cc = 0; cc < 32; ++cc) {
        float2 b0 = *reinterpret_cast<const float2*>(&wsh[cc * 128 + hh * 32 + ncol * 2]);
        float2 b1 = *reinterpret_cast<const float2*>(&wsh[cc * 128 + 64 + hh * 32 + ncol * 2]);
        v2f vb0 = {b0.x, b0.y}, vb1 = {b1.x, b1.y};
        v8f o = {0.f, 0.f, 0.f, 0.f, 0.f, 0.f, 0.f, 0.f};
        o = __builtin_amdgcn_wmma_f32_16x16x4_f32(false, va0, false, vb0, (short)0, o,
                                                  false, false);
        o = __builtin_amdgcn_wmma_f32_16x16x4_f32(false, va1, false, vb1, (short)0, o,
                                                  false, false);
        int c = cc * 16 + ncol;
        float boc = bos[c];
#pragma unroll
        for (int i = 0; i < 8; ++i) {
            int row = r0 + i + 8 * hh;
            float xv = xs[row * XS_STRIDE + c];
            float y = xv + SCALE_ * (o[i] + boc);
            xs[row * XS_STRIDE + c] = y;   // in-place y (x no longer needed here)
            rsum[i] += y;
            rsq[i]  += y * y;
        }
    }
    // reduce across the 16 lanes holding each row's columns (masks<16 stay in half)
#pragma unroll
    for (int i = 0; i < 8; ++i) {
        float s = rsum[i], q = rsq[i];
        for (int m = 1; m < 16; m <<= 1) {
            s += __shfl_xor(s, m);
            q += __shfl_xor(q, m);
        }
        float mu = s * (1.f / (float)DIM_);
        float var = q * (1.f / (float)DIM_) - mu * mu;  // biased, matches reference
        float rstd = rsqrtf(var + EPS_);
        if (ncol == 0) {
            mus[r0 + i + 8 * hh] = mu;
            rsds[r0 + i + 8 * hh] = rstd;
        }
    }
    __syncthreads();

    // ---- phase 3: normalize + fully-coalesced store
    float* outg = out + ((size_t)b * N_ + n0) * DIM_;
    for (int i = tid; i < ROWS_ * DIM_; i += 256) {
        int r = i >> 9, c = i & 511;
        float y = xs[r * XS_STRIDE + c];
        outg[(size_t)r * DIM_ + c] = gs[c] * (y - mus[r]) * rsds[r] + bs[c];
    }
}

extern "C" void kernel_launch(void* const* d_in, const int* in_sizes, int n_in,
                              void* d_out, int out_size, void* d_ws, size_t ws_size,
                              hipStream_t stream) {
    const float* x     = (const float*)d_in[0];
    const float* Wp    = (const float*)d_in[1];
    const float* bp    = (const float*)d_in[2];
    const float* Wm    = (const float*)d_in[3];
    const float* bm    = (const float*)d_in[4];
    const float* Wo    = (const float*)d_in[5];
    const float* bo    = (const float*)d_in[6];
    const float* gamma = (const float*)d_in[7];
    const float* beta  = (const float*)d_in[8];
    float* out = (float*)d_out;

    // workspace layout (floats): needs ~672KB
    float* ws      = (float*)d_ws;
    float* partial = ws;               // 8*32*512 = 131072
    float* mean    = ws + 131072;      // 8*512   = 4096
    float* t       = ws + 135168;      // 64
    float* W2      = ws + 135232;      // 8*8*512 = 32768

    k_partial<<<dim3(32, 8), 256, 0, stream>>>(x, partial);
    k_mean<<<16, 256, 0, stream>>>(partial, mean);
    k_t<<<1, 64, 0, stream>>>(mean, Wm, bm, t);
    k_w2<<<128, 256, 0, stream>>>(t, Wo, W2);

    const size_t lds_bytes = 78336 * sizeof(float);  // 313344 B < 320KB/WGP
    k_main<<<(B_ * N_) / ROWS_, 256, lds_bytes, stream>>>(x, Wp, bp, W2, bo, gamma, beta, out);
}